// HornerSparseIteration_LG_56040733278786
// MI455X (gfx1250) — compile-verified
//
#include <hip/hip_runtime.h>

#define NN      100000
#define CDIM    128
#define EPR     33        // edges per row (32 random + self loop, stably sorted)
#define T_ATTN  10
#define T_SP    10
#define CHUNK   512

typedef float v2f __attribute__((ext_vector_type(2)));
typedef float v8f __attribute__((ext_vector_type(8)));

// ---------------------------------------------------------------------------
// Projection: q = preds @ Wq^T + bq ; k = preds @ Wk^T + bk
// grid.x = NN/16 row tiles, 256 thr = 8 waves, wave w -> col tile 16*w
// ---------------------------------------------------------------------------
__global__ void proj_qk(const float* __restrict__ preds,
                        const float* __restrict__ Wq, const float* __restrict__ bq,
                        const float* __restrict__ Wk, const float* __restrict__ bk,
                        float* __restrict__ qo, float* __restrict__ ko) {
  const int lane = threadIdx.x & 31, wave = threadIdx.x >> 5;
  const int h = lane >> 4, m = lane & 15;
  const int m0 = blockIdx.x * 16;
  const int c0 = wave * 16;
  v8f accq = {}; v8f acck = {};
  for (int kk = 0; kk < CDIM; kk += 4) {
    v2f a  = *(const v2f*)(preds + (size_t)(m0 + m) * CDIM + kk + 2 * h);
    v2f b1 = *(const v2f*)(Wq    + (size_t)(c0 + m) * CDIM + kk + 2 * h);
    v2f b2 = *(const v2f*)(Wk    + (size_t)(c0 + m) * CDIM + kk + 2 * h);
    accq = __builtin_amdgcn_wmma_f32_16x16x4_f32(false, a, false, b1, (short)0, accq, false, false);
    acck = __builtin_amdgcn_wmma_f32_16x16x4_f32(false, a, false, b2, (short)0, acck, false, false);
  }
  const float biasq = bq[c0 + m];
  const float biask = bk[c0 + m];
#pragma unroll
  for (int v = 0; v < 8; ++v) {
    size_t o = (size_t)(m0 + v + 8 * h) * CDIM + c0 + m;
    qo[o] = accq[v] + biasq;
    ko[o] = acck[v] + biask;
  }
}

// ---------------------------------------------------------------------------
// Global sum-of-squares of q and k -> out2[0], out2[1]  (atomic accumulate)
// ---------------------------------------------------------------------------
__global__ void sumsq2(const float* __restrict__ q, const float* __restrict__ k,
                       float* __restrict__ out2, int count) {
  __shared__ float sA[256], sB[256];
  float a = 0.f, b = 0.f;
  for (int i = blockIdx.x * blockDim.x + threadIdx.x; i < count;
       i += gridDim.x * blockDim.x) {
    float x = q[i]; a = fmaf(x, x, a);
    float y = k[i]; b = fmaf(y, y, b);
  }
  sA[threadIdx.x] = a; sB[threadIdx.x] = b;
  __syncthreads();
  for (int s = 128; s > 0; s >>= 1) {
    if ((int)threadIdx.x < s) {
      sA[threadIdx.x] += sA[threadIdx.x + s];
      sB[threadIdx.x] += sB[threadIdx.x + s];
    }
    __syncthreads();
  }
  if (threadIdx.x == 0) { atomicAdd(&out2[0], sA[0]); atomicAdd(&out2[1], sB[0]); }
}

// ---------------------------------------------------------------------------
// Column sum of k (raw, scale folded in later). 128 blocks x 128 threads.
// ---------------------------------------------------------------------------
__global__ void colsum_k(const float* __restrict__ k, float* __restrict__ cs, int nrows) {
  const int c = threadIdx.x;
  const int rows_per = (nrows + (int)gridDim.x - 1) / (int)gridDim.x;
  int r0 = blockIdx.x * rows_per;
  int r1 = min(r0 + rows_per, nrows);
  float a = 0.f;
  for (int r = r0; r < r1; ++r) a += k[(size_t)r * CDIM + c];
  atomicAdd(&cs[c], a);
}

// ---------------------------------------------------------------------------
// attn_norm[n] = rq*rk*(q[n,:] . colsumK) + N   (one wave per row)
// ---------------------------------------------------------------------------
__global__ void attn_norm_k(const float* __restrict__ q, const float* __restrict__ cs,
                            const float* __restrict__ sqsk, float* __restrict__ an) {
  const int lane = threadIdx.x & 31;
  const int n = blockIdx.x * 8 + (threadIdx.x >> 5);
  float4 qa = ((const float4*)(q + (size_t)n * CDIM))[lane];
  float4 cv = ((const float4*)cs)[lane];
  float d = qa.x * cv.x + qa.y * cv.y + qa.z * cv.z + qa.w * cv.w;
  for (int o = 16; o; o >>= 1) d += __shfl_xor(d, o, 32);
  if (lane == 0) {
    float s = rsqrtf(sqsk[0]) * rsqrtf(sqsk[1]);
    an[n] = s * d + (float)NN;
  }
}

// ---------------------------------------------------------------------------
// dst = coef[ci] * src  (elementwise float4)
// ---------------------------------------------------------------------------
__global__ void scale_k(const float* __restrict__ src, float* __restrict__ dst,
                        const float* __restrict__ coef, int ci, int count4) {
  int i = blockIdx.x * blockDim.x + threadIdx.x;
  if (i < count4) {
    float c = coef[ci];
    float4 v = ((const float4*)src)[i];
    ((float4*)dst)[i] = make_float4(c * v.x, c * v.y, c * v.z, c * v.w);
  }
}

// ---------------------------------------------------------------------------
// Mout[d,c] += sum_n k[n,d]*X[n,c]  over chunk of rows (raw, unscaled).
// grid = (8 d-tiles, N/CHUNK chunks); wave w -> c tile 16*w; atomic accumulate.
// ---------------------------------------------------------------------------
__global__ void gemm_kT(const float* __restrict__ k, const float* __restrict__ X,
                        float* __restrict__ Mout, int nrows) {
  const int lane = threadIdx.x & 31, wave = threadIdx.x >> 5;
  const int h = lane >> 4, m = lane & 15;
  const int d0 = blockIdx.x * 16;
  const int c0 = wave * 16;
  const int n0 = blockIdx.y * CHUNK;
  const int n1 = min(n0 + CHUNK, nrows);
  v8f acc = {};
  for (int n = n0; n < n1; n += 4) {
    const int r0 = n + 2 * h;
    v2f a, b;
    a.x = k[(size_t)r0 * CDIM + d0 + m];
    a.y = k[(size_t)(r0 + 1) * CDIM + d0 + m];
    b.x = X[(size_t)r0 * CDIM + c0 + m];
    b.y = X[(size_t)(r0 + 1) * CDIM + c0 + m];
    acc = __builtin_amdgcn_wmma_f32_16x16x4_f32(false, a, false, b, (short)0, acc, false, false);
  }
#pragma unroll
  for (int v = 0; v < 8; ++v)
    atomicAdd(&Mout[(d0 + v + 8 * h) * CDIM + c0 + m], acc[v]);
}

// ---------------------------------------------------------------------------
// out[n,c] = s*(q@M)[n,c]/attn[n]  (+ coef[ci]*addBase[n,c] when addBase!=0)
// s = rq*rk = rsqrt(sq)*rsqrt(sk). grid.x = NN/16, wave w -> c tile 16*w.
// ---------------------------------------------------------------------------
__global__ void gemm_qM(const float* __restrict__ q, const float* __restrict__ M,
                        const float* __restrict__ attn, const float* __restrict__ sqsk,
                        const float* __restrict__ addBase, const float* __restrict__ coef,
                        int ci, float* __restrict__ out) {
  const int lane = threadIdx.x & 31, wave = threadIdx.x >> 5;
  const int h = lane >> 4, m = lane & 15;
  const int m0 = blockIdx.x * 16;
  const int c0 = wave * 16;
  v8f acc = {};
  for (int kk = 0; kk < CDIM; kk += 4) {
    v2f a = *(const v2f*)(q + (size_t)(m0 + m) * CDIM + kk + 2 * h);
    v2f b;
    b.x = M[(kk + 2 * h) * CDIM + c0 + m];
    b.y = M[(kk + 2 * h + 1) * CDIM + c0 + m];
    acc = __builtin_amdgcn_wmma_f32_16x16x4_f32(false, a, false, b, (short)0, acc, false, false);
  }
  const float s = rsqrtf(sqsk[0]) * rsqrtf(sqsk[1]);
  const float cf = addBase ? coef[ci] : 0.f;
#pragma unroll
  for (int v = 0; v < 8; ++v) {
    int row = m0 + v + 8 * h;
    float r = acc[v] * s / attn[row];
    size_t o = (size_t)row * CDIM + c0 + m;
    out[o] = addBase ? fmaf(cf, addBase[o], r) : r;
  }
}

// ---------------------------------------------------------------------------
// y[r,:] = coef[ci]*Hm[r,:] + sum_{e in [33r,33r+33)} val[e]*x[col[e],:]
// one wave per row; lane owns 4 columns (float4). Edge metadata scalarized.
// ---------------------------------------------------------------------------
__global__ void spmm_k(const float* __restrict__ vals, const int* __restrict__ cols,
                       const float* __restrict__ x, const float* __restrict__ Hm,
                       const float* __restrict__ coef, int ci, float* __restrict__ y) {
  const int lane = threadIdx.x & 31;
  const int r = blockIdx.x * 8 + (threadIdx.x >> 5);
  const int rr = __builtin_amdgcn_readfirstlane(r);
  const int e0 = rr * EPR;
  float4 acc = make_float4(0.f, 0.f, 0.f, 0.f);
  for (int e = 0; e < EPR; ++e) {
    float v = vals[e0 + e];
    int c = __builtin_amdgcn_readfirstlane(cols[e0 + e]);
    float4 xv = ((const float4*)(x + (size_t)c * CDIM))[lane];
    acc.x = fmaf(v, xv.x, acc.x);
    acc.y = fmaf(v, xv.y, acc.y);
    acc.z = fmaf(v, xv.z, acc.z);
    acc.w = fmaf(v, xv.w, acc.w);
  }
  const float a = coef[ci];
  float4 hv = ((const float4*)(Hm + (size_t)r * CDIM))[lane];
  ((float4*)(y + (size_t)r * CDIM))[lane] =
      make_float4(fmaf(a, hv.x, acc.x), fmaf(a, hv.y, acc.y),
                  fmaf(a, hv.z, acc.z), fmaf(a, hv.w, acc.w));
}

// ---------------------------------------------------------------------------
// out[i,:] = x[idx[i],:]   (2048 rows, one wave per row)
// ---------------------------------------------------------------------------
__global__ void gather_k(const float* __restrict__ x, const int* __restrict__ idx,
                         float* __restrict__ out, int nrows) {
  const int lane = threadIdx.x & 31;
  const int i = blockIdx.x * 8 + (threadIdx.x >> 5);
  if (i < nrows) {
    int n = idx[i];
    ((float4*)(out + (size_t)i * CDIM))[lane] =
        ((const float4*)(x + (size_t)n * CDIM))[lane];
  }
}

// ---------------------------------------------------------------------------
extern "C" void kernel_launch(void* const* d_in, const int* in_sizes, int n_in,
                              void* d_out, int out_size, void* d_ws, size_t ws_size,
                              hipStream_t stream) {
  const float* preds = (const float*)d_in[0];
  const float* Wq_w  = (const float*)d_in[1];
  const float* Wq_b  = (const float*)d_in[2];
  const float* Wk_w  = (const float*)d_in[3];
  const float* Wk_b  = (const float*)d_in[4];
  const float* beta  = (const float*)d_in[5];
  const float* alph  = (const float*)d_in[6];
  const float* evals = (const float*)d_in[7];
  const int*   ecol  = (const int*)d_in[9];
  const int*   idx   = (const int*)d_in[10];

  // workspace carve-up (all 512B-aligned; NC bytes = 51,200,000 = 512*100000)
  const size_t NC = (size_t)NN * CDIM * sizeof(float);
  char* w = (char*)d_ws;
  float* q   = (float*)(w);
  float* kb  = (float*)(w + NC);
  float* tmp = (float*)(w + 2 * NC);
  float* u   = (float*)(w + 3 * NC);
  char* sm = w + 4 * NC;
  float* an   = (float*)sm;                       // NN floats
  float* M1   = (float*)(sm + 400384);            // 128x128
  float* M2   = M1 + CDIM * CDIM;
  float* cs   = M2 + CDIM * CDIM;                 // 128 colsums
  float* sqsk = cs + CDIM;                        // 2 scalars

  const int rowBlocks = NN / 16;                  // 6250
  const int waveRowBlocks = NN / 8;               // 12500 (8 waves/block)
  const int nChunks = (NN + CHUNK - 1) / CHUNK;   // 196

  // zero atomic accumulators (colsum + sumsq scalars, contiguous)
  hipMemsetAsync(cs, 0, (CDIM + 2) * sizeof(float), stream);

  // phase 1: projections, norms, attn_norm, tmp init
  proj_qk<<<rowBlocks, 256, 0, stream>>>(preds, Wq_w, Wq_b, Wk_w, Wk_b, q, kb);
  sumsq2<<<1024, 256, 0, stream>>>(q, kb, sqsk, NN * CDIM);
  colsum_k<<<128, 128, 0, stream>>>(kb, cs, NN);
  attn_norm_k<<<waveRowBlocks, 256, 0, stream>>>(q, cs, sqsk, an);
  scale_k<<<(NN * CDIM / 4 + 255) / 256, 256, 0, stream>>>(preds, tmp, beta, T_ATTN - 1,
                                                           NN * CDIM / 4);

  // phase 2: attention Horner loop (Hmat == preds throughout)
  for (int i = T_ATTN - 1; i >= 1; --i) {
    hipMemsetAsync(M1, 0, CDIM * CDIM * sizeof(float), stream);
    gemm_kT<<<dim3(CDIM / 16, nChunks), 256, 0, stream>>>(kb, tmp, M1, NN);
    gemm_qM<<<rowBlocks, 256, 0, stream>>>(q, M1, an, sqsk, nullptr, beta, 0, u);
    hipMemsetAsync(M2, 0, CDIM * CDIM * sizeof(float), stream);
    gemm_kT<<<dim3(CDIM / 16, nChunks), 256, 0, stream>>>(kb, u, M2, NN);
    gemm_qM<<<rowBlocks, 256, 0, stream>>>(q, M2, an, sqsk, preds, beta, i - 1, tmp);
  }

  // phase 3: sparse Horner loop. Hmat = tmp; ping-pong x/y through u and q.
  scale_k<<<(NN * CDIM / 4 + 255) / 256, 256, 0, stream>>>(tmp, u, alph, T_SP - 1,
                                                           NN * CDIM / 4);
  float* x = u;
  float* y = q;
  for (int i = T_SP - 1; i >= 1; --i) {
    spmm_k<<<waveRowBlocks, 256, 0, stream>>>(evals, ecol, x, tmp, alph, i - 1, y);
    float* t = x; x = y; y = t;
  }

  // phase 4: gather 2048 rows
  gather_k<<<(2048 + 7) / 8, 256, 0, stream>>>(x, idx, (float*)d_out, 2048);
}